// TopKGating_85023172591905
// MI455X (gfx1250) — compile-verified
//
#include <hip/hip_runtime.h>
#include <cmath>
#include <stdint.h>

typedef __attribute__((ext_vector_type(2))) float v2f;
typedef __attribute__((ext_vector_type(8))) float v8f;

#define NTOK 16384
#define DIM  4096
#define NEXP 64
#define TOPK 8
#define KT   64                       // K tile
#define SWS  66                       // sW row stride: bank-pair-safe b64 reads
#define SAS  68                       // sA row stride: 16B-aligned, stride-4 banks
#define TPB  128                      // tokens per block (8 waves x 16 rows)
#define SW_FLOATS (NEXP * SWS)        // 4224
#define SA_FLOATS (8 * 16 * SAS)      // 8704
#define BUF_FLOATS (SW_FLOATS + SA_FLOATS)  // 12928 floats per buffer
#define NCH  (DIM / KT)               // 64 K-chunks

__global__ __launch_bounds__(256) void
moe_topk_gate(const float* __restrict__ x, const float* __restrict__ w,
              int* __restrict__ out_idx, float* __restrict__ out_gate)
{
    __shared__ __align__(16) float smem[2 * BUF_FLOATS];  // ~103 KB, double-buffered

    const int tid  = threadIdx.x;
    const int wave = tid >> 5;
    const int lane = tid & 31;
    const int m    = lane & 15;       // M row (A/C) / N col (B)
    const int g    = lane >> 4;       // half-wave -> K-pair select
    const int tok0 = blockIdx.x * TPB;

    // per-thread w-staging coordinates
    const int krow = tid >> 4;        // 0..15
    const int e4   = tid & 15;        // experts e4*4..e4*4+3

    v8f c0 = {}, c1 = {}, c2 = {}, c3 = {};

    // ================= prologue: fetch chunk 0 into buffer 0 =================
    {
        // async x tile -> sA(buf0): per-lane LDS scatter keeps the padded layout
#pragma unroll
        for (int i = 0; i < 8; ++i) {
            const int f4 = i * 32 + lane, row = f4 >> 4, c4 = f4 & 15;
            const uint32_t lds = (uint32_t)((SW_FLOATS + (wave * 16 + row) * SAS + c4 * 4) * 4);
            const int voff = (((tok0 + wave * 16 + row) * DIM) + c4 * 4) * 4;   // bytes, < 2^31
            asm volatile("global_load_async_to_lds_b128 %0, %1, %2"
                         :: "v"(lds), "v"(voff), "s"(x) : "memory");
        }
        // w chunk 0 -> regs -> transposed sW(buf0)
        float4 wv[4];
#pragma unroll
        for (int i = 0; i < 4; ++i)
            wv[i] = *(const float4*)&w[(size_t)(krow + i * 16) * NEXP + e4 * 4];
        float* sW0 = smem;
#pragma unroll
        for (int i = 0; i < 4; ++i) {
            const int kk = krow + i * 16;
            sW0[(e4 * 4 + 0) * SWS + kk] = wv[i].x;
            sW0[(e4 * 4 + 1) * SWS + kk] = wv[i].y;
            sW0[(e4 * 4 + 2) * SWS + kk] = wv[i].z;
            sW0[(e4 * 4 + 3) * SWS + kk] = wv[i].w;
        }
        asm volatile("s_wait_asynccnt 0x0" ::: "memory");
        __syncthreads();
    }

    // ================= main pipelined loop: one barrier per chunk =================
    for (int kc = 0; kc < NCH; ++kc) {
        const int p = kc & 1;
        const int n = p ^ 1;
        const bool more = (kc + 1) < NCH;
        float* sWp = smem + p * BUF_FLOATS;
        float* sAp = sWp + SW_FLOATS;

        float4 wv[4];
        if (more) {
            const int knext = (kc + 1) * KT;
            // prefetch next x tile async -> sA(buf n); lands while we do WMMAs
#pragma unroll
            for (int i = 0; i < 8; ++i) {
                const int f4 = i * 32 + lane, row = f4 >> 4, c4 = f4 & 15;
                const uint32_t lds =
                    (uint32_t)((n * BUF_FLOATS + SW_FLOATS + (wave * 16 + row) * SAS + c4 * 4) * 4);
                const int voff = (((tok0 + wave * 16 + row) * DIM) + knext + c4 * 4) * 4;
                asm volatile("global_load_async_to_lds_b128 %0, %1, %2"
                             :: "v"(lds), "v"(voff), "s"(x) : "memory");
            }
            // prefetch next w chunk into regs (L2-resident, loadcnt path)
#pragma unroll
            for (int i = 0; i < 4; ++i)
                wv[i] = *(const float4*)&w[(size_t)(knext + krow + i * 16) * NEXP + e4 * 4];
        }

        // ---- compute on buffer p: 16 K-steps x 4 expert tiles of WMMA f32 16x16x4
#pragma unroll
        for (int k = 0; k < KT; k += 4) {
            v2f a  = *(const v2f*)&sAp[(wave * 16 + m) * SAS + k + 2 * g];
            v2f b0 = *(const v2f*)&sWp[( 0 + m) * SWS + k + 2 * g];
            v2f b1 = *(const v2f*)&sWp[(16 + m) * SWS + k + 2 * g];
            v2f b2 = *(const v2f*)&sWp[(32 + m) * SWS + k + 2 * g];
            v2f b3 = *(const v2f*)&sWp[(48 + m) * SWS + k + 2 * g];
            c0 = __builtin_amdgcn_wmma_f32_16x16x4_f32(false, a, false, b0, (short)0, c0, false, false);
            c1 = __builtin_amdgcn_wmma_f32_16x16x4_f32(false, a, false, b1, (short)0, c1, false, false);
            c2 = __builtin_amdgcn_wmma_f32_16x16x4_f32(false, a, false, b2, (short)0, c2, false, false);
            c3 = __builtin_amdgcn_wmma_f32_16x16x4_f32(false, a, false, b3, (short)0, c3, false, false);
        }

        if (more) {
            // transposed store of prefetched w regs into sW(buf n)
            float* sWn = smem + n * BUF_FLOATS;
#pragma unroll
            for (int i = 0; i < 4; ++i) {
                const int kk = krow + i * 16;
                sWn[(e4 * 4 + 0) * SWS + kk] = wv[i].x;
                sWn[(e4 * 4 + 1) * SWS + kk] = wv[i].y;
                sWn[(e4 * 4 + 2) * SWS + kk] = wv[i].z;
                sWn[(e4 * 4 + 3) * SWS + kk] = wv[i].w;
            }
            asm volatile("s_wait_asynccnt 0x0" ::: "memory");  // x tile for buf n landed
        }
        __syncthreads();  // orders this chunk's reads before next chunk's writes
    }

    // ================= epilogue: logits -> LDS, softmax + top-8 =================
    float* sL = smem;   // 128*66 = 8448 floats; all GEMM LDS reads retired at last barrier
#pragma unroll
    for (int r = 0; r < 8; ++r) {
        const int row = wave * 16 + g * 8 + r;  // C layout: lanes 0-15 -> M=r, 16-31 -> M=r+8
        sL[row * SWS +  0 + m] = c0[r];
        sL[row * SWS + 16 + m] = c1[r];
        sL[row * SWS + 32 + m] = c2[r];
        sL[row * SWS + 48 + m] = c3[r];
    }
    __syncthreads();

    if (tid < TPB) {
        float* rowp = &sL[tid * SWS];
        float mx = -INFINITY;
        for (int e = 0; e < NEXP; ++e) mx = fmaxf(mx, rowp[e]);
        float s = 0.0f;
        for (int e = 0; e < NEXP; ++e) s += __expf(rowp[e] - mx);
        const float inv = 1.0f / s;
        const int tok = tok0 + tid;
#pragma unroll
        for (int j = 0; j < TOPK; ++j) {
            float best = -INFINITY; int bi = 0;
            for (int e = 0; e < NEXP; ++e) {
                float v = rowp[e];
                if (v > best) { best = v; bi = e; }   // strict '>' == lax.top_k tie-break
            }
            rowp[bi] = -INFINITY;
            out_idx [tok * TOPK + j] = bi;
            out_gate[tok * TOPK + j] = __expf(best - mx) * inv;
        }
    }
}

extern "C" void kernel_launch(void* const* d_in, const int* in_sizes, int n_in,
                              void* d_out, int out_size, void* d_ws, size_t ws_size,
                              hipStream_t stream) {
    (void)in_sizes; (void)n_in; (void)d_ws; (void)ws_size; (void)out_size;
    const float* x = (const float*)d_in[0];        // [16384, 4096] fp32
    const float* w = (const float*)d_in[1];        // [4096, 64]    fp32
    // tuple output: (top_k_indices int32, top_k_gates fp32), concatenated flat
    int*   out_idx  = (int*)d_out;                          // first 16384*8 elements
    float* out_gate = (float*)d_out + (size_t)NTOK * TOPK;  // next  16384*8 elements

    dim3 grid(NTOK / TPB);   // 128 blocks
    dim3 block(256);         // 8 waves (wave32)
    moe_topk_gate<<<grid, block, 0, stream>>>(x, w, out_idx, out_gate);
}